// DGATModel_11304353923835
// MI455X (gfx1250) — compile-verified
//
#include <hip/hip_runtime.h>
#include <math.h>

typedef __attribute__((ext_vector_type(2))) float v2f;
typedef __attribute__((ext_vector_type(8))) float v8f;

#define ALPHA 0.2f
#define FDIM  128        // F_in == F_out == 128
#define DNB   16         // neighbors per node
#define LSEQ  4096       // sequence length per batch
#define BL    32768      // B * L rows

// ---------------------------------------------------------------------------
// Kernel 1: h = x @ W   (32768x128 @ 128x128, f32) via V_WMMA_F32_16X16X4_F32
// 256 threads = 8 waves per block; each wave computes a 16x128 output stripe.
// W is staged transposed in LDS so every B fragment is one ds_load_b64.
// ---------------------------------------------------------------------------
__global__ __launch_bounds__(256)
void gat_gemm_xW(const float* __restrict__ x, const float* __restrict__ W,
                 float* __restrict__ h) {
  __shared__ float Wt[FDIM * FDIM];   // Wt[n*128 + k] = W[k*128 + n]  (64 KB)

  for (int i = threadIdx.x; i < FDIM * FDIM; i += blockDim.x) {
    int k = i >> 7, n = i & 127;
    Wt[n * FDIM + k] = W[i];
  }
  __syncthreads();

  const int wave  = threadIdx.x >> 5;
  const int lane  = threadIdx.x & 31;
  const int row0  = (blockIdx.x * 8 + wave) * 16;   // 16 output rows per wave
  const int m     = lane & 15;                      // A row / B,N column slot
  const int khalf = (lane >> 4) << 1;               // 0 or 2 (K sub-pair)

  // Preload the entire 16x128 A tile as 32 K-step fragments (vmem clause).
  // ISA A-layout (32-bit 16x4): lanes 0-15 hold K=k0..k0+1, lanes 16-31 K=k0+2..k0+3.
  v2f afrag[32];
  const float* xrow = x + (size_t)(row0 + m) * FDIM + khalf;
  #pragma unroll
  for (int ks = 0; ks < 32; ++ks)
    afrag[ks] = *(const v2f*)(xrow + ks * 4);

  v8f acc[8] = {};   // 8 N-tiles of 16x16 f32 accumulators

  #pragma unroll
  for (int ks = 0; ks < 32; ++ks) {
    const int kb = ks * 4 + khalf;
    #pragma unroll
    for (int n = 0; n < 8; ++n) {
      // B-layout (4x16): lanes 0-15 hold K=k0..k0+1 of column n*16+m, lanes 16-31 K=k0+2..k0+3
      v2f bfrag = *(const v2f*)(&Wt[(n * 16 + m) * FDIM + kb]);
      acc[n] = __builtin_amdgcn_wmma_f32_16x16x4_f32(
          /*neg_a=*/false, afrag[ks], /*neg_b=*/false, bfrag,
          /*c_mod=*/(short)0, acc[n], /*reuse_a=*/false, /*reuse_b=*/false);
    }
  }

  // D layout: VGPR r, lane l -> row = r + (l>>4)*8, col = n*16 + (l&15)
  const int mhi = (lane >> 4) << 3;
  #pragma unroll
  for (int n = 0; n < 8; ++n) {
    #pragma unroll
    for (int r = 0; r < 8; ++r) {
      h[(size_t)(row0 + r + mhi) * FDIM + n * 16 + m] = acc[n][r];
    }
  }
}

// ---------------------------------------------------------------------------
// Kernel 2: gather + attention + softmax + aggregate + ELU.
// One wave32 per node; lane owns 4 features -> 512B coalesced row loads (L2 hits).
// ---------------------------------------------------------------------------
__device__ __forceinline__ float wave_sum(float s) {
  #pragma unroll
  for (int off = 16; off > 0; off >>= 1)
    s += __shfl_xor(s, off, 32);
  return s;
}

__global__ __launch_bounds__(256)
void gat_attn(const float* __restrict__ h, const int* __restrict__ adj,
              const float* __restrict__ a, float* __restrict__ out) {
  const int wid  = (int)((blockIdx.x * blockDim.x + threadIdx.x) >> 5);
  const int lane = threadIdx.x & 31;
  if (wid >= BL) return;

  const int    b    = wid >> 12;          // wid / 4096
  const size_t base = (size_t)b * LSEQ;   // batch row offset into h
  const float4* hv  = (const float4*)h;

  // Gather 16 neighbor rows; each lane keeps its 4 features of every row.
  float4 v[DNB];
  const int* arow = adj + (size_t)wid * DNB;
  #pragma unroll
  for (int d = 0; d < DNB; ++d) {
    const int nb = arow[d];
    v[d] = hv[(base + (size_t)nb) * (FDIM / 4) + lane];
  }

  const float4 alo = ((const float4*)a)[lane];        // a[0:128]
  const float4 ahi = ((const float4*)a)[32 + lane];   // a[128:256]

  // e_hi = h_prime[:,0] . a_hi  (shared across all D)
  const float ehi = wave_sum(v[0].x * ahi.x + v[0].y * ahi.y +
                             v[0].z * ahi.z + v[0].w * ahi.w);

  float e[DNB];
  float emax = -1e30f;
  #pragma unroll
  for (int d = 0; d < DNB; ++d) {
    float s = wave_sum(v[d].x * alo.x + v[d].y * alo.y +
                       v[d].z * alo.z + v[d].w * alo.w) + ehi;
    s = (s > 0.f) ? s : ALPHA * s;    // LeakyReLU
    e[d] = s;
    emax = fmaxf(emax, s);
  }

  float denom = 0.f;
  #pragma unroll
  for (int d = 0; d < DNB; ++d) { e[d] = __expf(e[d] - emax); denom += e[d]; }
  const float inv = 1.f / denom;

  float4 o = {0.f, 0.f, 0.f, 0.f};
  #pragma unroll
  for (int d = 0; d < DNB; ++d) {
    const float w = e[d] * inv;
    o.x += w * v[d].x; o.y += w * v[d].y;
    o.z += w * v[d].z; o.w += w * v[d].w;
  }

  // ELU
  o.x = (o.x > 0.f) ? o.x : (__expf(o.x) - 1.f);
  o.y = (o.y > 0.f) ? o.y : (__expf(o.y) - 1.f);
  o.z = (o.z > 0.f) ? o.z : (__expf(o.z) - 1.f);
  o.w = (o.w > 0.f) ? o.w : (__expf(o.w) - 1.f);

  ((float4*)out)[(size_t)wid * (FDIM / 4) + lane] = o;
}

// ---------------------------------------------------------------------------
extern "C" void kernel_launch(void* const* d_in, const int* in_sizes, int n_in,
                              void* d_out, int out_size, void* d_ws, size_t ws_size,
                              hipStream_t stream) {
  const float* x   = (const float*)d_in[0];   // [8,4096,128] f32
  const int*   adj = (const int*)d_in[1];     // [8,4096,16]  int
  const float* W   = (const float*)d_in[2];   // [128,128]    f32
  const float* a   = (const float*)d_in[3];   // [256,1]      f32
  float*       out = (float*)d_out;           // [8,4096,128] f32
  float*       h   = (float*)d_ws;            // scratch: 32768*128 f32 = 16 MB

  // Kernel 1: 2048 wave-tiles of 16 rows, 8 waves per block -> 256 blocks.
  gat_gemm_xW<<<256, 256, 0, stream>>>(x, W, h);
  // Kernel 2: 32768 nodes, one wave each, 8 waves per block -> 4096 blocks.
  gat_attn<<<4096, 256, 0, stream>>>(h, adj, a, out);
}